// DynamicImpactEncoder_82583631167653
// MI455X (gfx1250) — compile-verified
//
#include <hip/hip_runtime.h>

// ---------------------------------------------------------------------------
// impact[b,t,c] = alpha[c] * y[b,t,c],  y[t] = x[t] + r*y[t+1],  r = e^(-1/decay)
// Memory-bound (AI ~ 0.4 flop/byte): single pass, 96MiB in + 96MiB out,
// ~8.6us floor at 23.3 TB/s. Intra-chunk scan = triangular matmul on
// v_wmma_f32_16x16x4_f32; inter-chunk scan = wave-parallel shuffle scan;
// global<->LDS staging double-buffered via async LDS copies (ASYNCcnt).
// ---------------------------------------------------------------------------

typedef float v2f __attribute__((ext_vector_type(2)));
typedef float v4f __attribute__((ext_vector_type(4)));
typedef float v8f __attribute__((ext_vector_type(8)));

#define T_LEN   16384
#define C_CH    3
#define SEG     2048                 // timesteps per LDS segment
#define NSEG    (T_LEN / SEG)        // 8
#define CHUNK   16
#define TILE_T  256                  // 16 chunks of 16 = one WMMA tile
#define NTILE   (SEG / TILE_T)       // 8 tiles per channel (== NWAVE)
#define NCHUNK  (SEG / CHUNK)        // 128 chunks per channel
#define CPL     (NCHUNK / 32)        // 4 chunks per lane in the scan
#define BLOCK   256
#define NWAVE   (BLOCK / 32)         // 8 wave32s
#define TPW     C_CH                 // tasks per wave: c = tk, tile = wave
#define SEGF    (SEG * C_CH)         // 6144 floats per segment
#define LPT     (SEGF / 4 / BLOCK)   // 6 b128 transfers per thread

// Builtin presence implies the toolchain knows the gfx1250 async-LDS ISA;
// we emit the instructions via inline asm (the builtin's pointer parameter
// types use the un-spellable __device__ LangAS in HIP).
#if __has_builtin(__builtin_amdgcn_global_load_async_to_lds_b128) && \
    __has_builtin(__builtin_amdgcn_global_store_async_from_lds_b128)
#define HAVE_ASYNC 1
#else
#define HAVE_ASYNC 0
#warning "gfx1250 async LDS path unavailable; using synchronous copies"
#endif

__device__ __forceinline__ unsigned lds_addr_of(const void* p) {
    // generic -> LDS(AS3) addrspacecast, then to 32-bit LDS byte address
    return (unsigned)(unsigned long long)
           (__attribute__((address_space(3))) const void*)p;
}

__device__ __forceinline__ void async_load_b128(const void* gsrc, void* ldst) {
#if HAVE_ASYNC
    asm volatile("global_load_async_to_lds_b128 %0, %1, off"
                 :: "v"(lds_addr_of(ldst)), "v"(gsrc) : "memory");
#endif
}

__device__ __forceinline__ void async_store_b128(void* gdst, const void* lsrc) {
#if HAVE_ASYNC
    asm volatile("global_store_async_from_lds_b128 %0, %1, off"
                 :: "v"(gdst), "v"(lds_addr_of(lsrc)) : "memory");
#endif
}

__device__ __forceinline__ void wait_async_all() {
#if HAVE_ASYNC
    asm volatile("s_wait_asynccnt 0x0" ::: "memory");
#endif
}

__launch_bounds__(BLOCK)
__global__ void DynamicImpactEncoder_kernel(const float* __restrict__ events,
                                            const float* __restrict__ decay,
                                            const float* __restrict__ alpha,
                                            float* __restrict__ out)
{
    __shared__ __align__(16) float stage[2][SEGF];      // 2 x 24KB staging
    __shared__ v2f   bstage[C_CH][4][32];               // B fragments (c,kb,lane)
    __shared__ float s_lds[C_CH][NCHUNK];               // chunk-start partials
    __shared__ float carr [C_CH][NCHUNK];               // carry INTO each chunk
    __shared__ float rtab [C_CH][CHUNK + 1];            // r^0 .. r^16
    __shared__ float carry_seg[C_CH];                   // y at start of next segment

    const int tid  = threadIdx.x;
    const int b    = blockIdx.x;
    const int lane = tid & 31;
    const int wave = tid >> 5;
    const int half = lane >> 4;     // 0: lanes 0-15, 1: lanes 16-31
    const int lml  = lane & 15;     // M for A-frag, N for B-frag and C/D

    // ---- one-time init ----------------------------------------------------
    if (tid < C_CH * (CHUNK + 1)) {
        int c = tid / (CHUNK + 1), e = tid % (CHUNK + 1);
        rtab[c][e] = __expf(-(float)e / decay[c]);      // r^e
    }
    if (tid < C_CH) carry_seg[tid] = 0.0f;

    float r16_c = 0.f, f0_c = 0.f, fpow_c = 0.f;
    if (wave < C_CH) {
        const float nid = -1.0f / decay[wave];
        r16_c  = __expf(16.0f * nid);                         // r^16
        f0_c   = __expf(16.0f * (float)CPL * nid);            // r^64
        fpow_c = __expf(16.0f * (float)CPL * (float)(31 - lane) * nid);
        // B[j][n] = r^(j-n) for j>=n else 0;  lane: n=lml, K pair j0/j0+1.
        #pragma unroll
        for (int kb = 0; kb < 4; ++kb) {
            const int j0 = 4 * kb + 2 * half;
            const int e0 = j0 - lml, e1 = e0 + 1;
            v2f bv;
            bv.x = (e0 >= 0) ? __expf((float)e0 * nid) : 0.0f;
            bv.y = (e1 >= 0) ? __expf((float)e1 * nid) : 0.0f;
            bstage[wave][kb][lane] = bv;
        }
    }
    const float a0 = alpha[0], a1 = alpha[1], a2 = alpha[2];
    const float alph[3] = {a0, a1, a2};

    const size_t row0 = (size_t)b * T_LEN * C_CH;

    // ---- copy helpers -----------------------------------------------------
    auto issue_load = [&](float* buf, size_t segf) {
        #pragma unroll
        for (int k = 0; k < LPT; ++k) {
            const int idx = tid + k * BLOCK;            // b128 index
#if HAVE_ASYNC
            async_load_b128(events + segf + (size_t)idx * 4, buf + idx * 4);
#else
            ((v4f*)buf)[idx] = ((const v4f*)(events + segf))[idx];
#endif
        }
    };
    auto issue_store = [&](const float* buf, size_t segf) {
        #pragma unroll
        for (int k = 0; k < LPT; ++k) {
            const int idx = tid + k * BLOCK;
#if HAVE_ASYNC
            async_store_b128(out + segf + (size_t)idx * 4, buf + idx * 4);
#else
            ((v4f*)(out + segf))[idx] = ((const v4f*)buf)[idx];
#endif
        }
    };

    // ---- software pipeline over segments (backward) -----------------------
    int cur = 0;
    issue_load(stage[0], row0 + (size_t)(NSEG - 1) * SEGF);

    for (int i = 0; i < NSEG; ++i) {
        const int seg = NSEG - 1 - i;
        const size_t segf = row0 + (size_t)seg * SEGF;

        wait_async_all();            // cur's load done; cur's old store done
        __syncthreads();             // every wave's asyncs retired

        if (seg > 0)                 // prefetch next segment into other buffer
            issue_load(stage[1 - cur], segf - SEGF);

        float* const st = stage[cur];

        // ---- WMMA: P[q][k] = sum_{j>=k} r^{j-k} X[q][j], tile = wave ------
        v8f acc[TPW];
        #pragma unroll
        for (int tk = 0; tk < TPW; ++tk) {
            const int c     = tk;            // compile-time channel
            const int tbase = wave * TILE_T;
            v8f d = {};
            #pragma unroll
            for (int kb = 0; kb < 4; ++kb) {
                const int j0 = 4 * kb + 2 * half;
                v2f av;                       // A: row M=lml (16x4 layout)
                av.x = st[(tbase + 16 * lml + j0    ) * C_CH + c];
                av.y = st[(tbase + 16 * lml + j0 + 1) * C_CH + c];
                const v2f bv = bstage[c][kb][lane];
                d = __builtin_amdgcn_wmma_f32_16x16x4_f32(
                        false, av, false, bv, (short)0, d, false, false);
            }
            acc[tk] = d;
            if (lml == 0) {                   // column N==0: chunk-start values
                #pragma unroll
                for (int v = 0; v < 8; ++v)
                    s_lds[c][wave * CHUNK + 8 * half + v] = acc[tk][v];
            }
        }
        __syncthreads();

        // ---- wave-parallel inter-chunk scan (waves 0..2, c = wave) --------
        // G[q] = s[q] + r16*G[q+1], G[128] = carry_seg;  carr[q] = G[q+1].
        if (wave < C_CH) {
            const int c = wave;
            float sloc[CPL];
            #pragma unroll
            for (int k = 0; k < CPL; ++k) sloc[k] = s_lds[c][lane * CPL + k];

            float S = 0.0f;                                  // local decayed sum
            #pragma unroll
            for (int k = CPL - 1; k >= 0; --k) S = sloc[k] + r16_c * S;

            float T = S, fd = f0_c;                          // backward scan
            #pragma unroll
            for (int dlt = 1; dlt < 32; dlt <<= 1) {
                float o = __shfl_down(T, dlt, 32);
                if (lane + dlt < 32) T += fd * o;
                fd *= fd;
            }
            float Tn = __shfl_down(T, 1, 32);
            if (lane == 31) Tn = 0.0f;
            const float cseg = carry_seg[c];
            float y = Tn + fpow_c * cseg;                    // G[(lane+1)*CPL]
            #pragma unroll
            for (int k = CPL - 1; k >= 0; --k) {
                carr[c][lane * CPL + k] = y;
                y = sloc[k] + r16_c * y;
            }
            if (lane == 0) carry_seg[c] = T + f0_c * fpow_c * cseg;  // G[0]
        }
        __syncthreads();

        // ---- combine + scale, write back into current staging buffer -----
        #pragma unroll
        for (int tk = 0; tk < TPW; ++tk) {
            const int c = tk;
            const float ac = alph[c];
            const float rk = rtab[c][CHUNK - lml];           // r^(16-k)
            #pragma unroll
            for (int v = 0; v < 8; ++v) {
                const int q  = v + 8 * half;                 // chunk within tile
                const int tl = wave * TILE_T + 16 * q + lml;
                const float y = acc[tk][v] + rk * carr[c][wave * CHUNK + q];
                st[tl * C_CH + c] = ac * y;
            }
        }
        __syncthreads();

        issue_store(st, segf);       // async DMA LDS -> global
        cur ^= 1;
    }
    wait_async_all();                // drain before endpgm (also implicit)
}

extern "C" void kernel_launch(void* const* d_in, const int* in_sizes, int n_in,
                              void* d_out, int out_size, void* d_ws, size_t ws_size,
                              hipStream_t stream)
{
    const float* events = (const float*)d_in[0];
    const float* decay  = (const float*)d_in[1];
    const float* alpha  = (const float*)d_in[2];
    float* out = (float*)d_out;

    const int B = in_sizes[0] / (T_LEN * C_CH);   // 512
    DynamicImpactEncoder_kernel<<<B, BLOCK, 0, stream>>>(events, decay, alpha, out);
}